// HomogeneousCropEfficient_64029372448826
// MI455X (gfx1250) — compile-verified
//
#include <hip/hip_runtime.h>

// ---------------------------------------------------------------------------
// HomogeneousCropEfficient for MI455X (gfx1250, wave32)
//
// Stage A: streaming block sums, float4/global_load_b128 per lane
//          (288 MB read once -> ~12.4 us @ 23.3 TB/s; vmem-issue minimized)
// Stage B: 8x8 box filter over block sums as banded-ones matmul on the
//          fp32 WMMA pipe (v_wmma_f32_16x16x4_f32)
// Stage C: argmin of variance score (first-index tie-break, like jnp.argmin)
// Stage D: vectorized crop copy
// ---------------------------------------------------------------------------

typedef __attribute__((ext_vector_type(2))) float v2f;
typedef __attribute__((ext_vector_type(8))) float v8f;

#define IMG_C   3
#define IMG_H   4000
#define IMG_W   6000
#define PLANE   (IMG_H * IMG_W)      // 24,000,000
#define PATCH   512
#define STEP    64
#define NH      55                   // h corners: 0,64,...,3456
#define NW      86                   // w corners: 0,64,...,5440
#define NQ      62                   // row chunks q: rows 64q+1 .. 64q+64
#define NB      93                   // col blocks b: cols 64b   .. 64b+63

// ---------------------------------------------------------------------------
// Stage A: one workgroup per (q,b) cell of the 62x93 chunk grid.
// Each cell sums gray and gray^2 over a 64x64 region (rows shifted by +1 to
// match the reference's padded-integral-image box).
// Thread map: lane covers 4 consecutive cols (float4), 16 lanes span the 64
// cols of one row; 256 threads cover 16 rows per iteration, 4 iterations.
// ---------------------------------------------------------------------------
__global__ void __launch_bounds__(256)
stageA_blocksums(const float* __restrict__ t,
                 float* __restrict__ c1, float* __restrict__ c2) {
    const int b    = blockIdx.x;           // 0..92
    const int q    = blockIdx.y;           // 0..61
    const int tid  = threadIdx.x;          // 0..255
    const int c4   = (tid & 15) * 4;       // col offset in block (0,4,...,60)
    const int rsub = tid >> 4;             // 0..15
    const int col  = b * 64 + c4;
    const int r0   = q * 64 + 1 + rsub;

    const float* p0 = t + (size_t)r0 * IMG_W + col;   // channel 0, 16B aligned

    float s1 = 0.0f, s2 = 0.0f;
#pragma unroll
    for (int it = 0; it < 4; ++it) {
        const size_t off = (size_t)(it * 16) * IMG_W;
        // prefetch 16 rows ahead (speculative; gfx1250 global_prefetch_b8)
        __builtin_prefetch(p0 + off + (size_t)16 * IMG_W, 0, 0);
        __builtin_prefetch(p0 + off + (size_t)16 * IMG_W + PLANE, 0, 0);
        __builtin_prefetch(p0 + off + (size_t)16 * IMG_W + 2 * (size_t)PLANE, 0, 0);

        const float4 a  = *reinterpret_cast<const float4*>(p0 + off);
        const float4 bb = *reinterpret_cast<const float4*>(p0 + off + PLANE);
        const float4 cc = *reinterpret_cast<const float4*>(p0 + off + 2 * (size_t)PLANE);

        const float g0 = (a.x + bb.x + cc.x) * (1.0f / 3.0f);
        const float g1 = (a.y + bb.y + cc.y) * (1.0f / 3.0f);
        const float g2 = (a.z + bb.z + cc.z) * (1.0f / 3.0f);
        const float g3 = (a.w + bb.w + cc.w) * (1.0f / 3.0f);

        s1 += (g0 + g1) + (g2 + g3);
        s2 += (g0 * g0 + g1 * g1) + (g2 * g2 + g3 * g3);
    }

    __shared__ float r1[256];
    __shared__ float r2[256];
    r1[tid] = s1;
    r2[tid] = s2;
    __syncthreads();
#pragma unroll
    for (int s = 128; s > 0; s >>= 1) {
        if (tid < s) {
            r1[tid] += r1[tid + s];
            r2[tid] += r2[tid + s];
        }
        __syncthreads();
    }
    if (tid == 0) {
        c1[q * NB + b] = r1[0];
        c2[q * NB + b] = r2[0];
    }
}

// ---------------------------------------------------------------------------
// Stage B: T[i,j] = sum_{dq<8, db<8} C[i+dq, j+db]  as  T = W * H
//   H[k,n] = sum_{db<8} C[k, n+db]      (computed on the fly)
//   W[i,k] = 1 if i <= k <= i+7 else 0  (banded ones, generated in registers)
// Run on v_wmma_f32_16x16x4_f32: full fp32 precision on the matrix pipe.
// One wave per 16x16 output tile per plane. blockDim.x == 32 -> EXEC all 1s
// at the WMMA (guarded loads reconverge before issue).
// ---------------------------------------------------------------------------
__device__ __forceinline__ float hval(const float* __restrict__ Cp, int k, int n) {
    if (k >= NQ || n >= NW) return 0.0f;   // OOB -> contributes zero
    const float* row = Cp + k * NB + n;    // n+7 <= 85+7 = 92 = NB-1, in range
    float s = 0.0f;
#pragma unroll
    for (int db = 0; db < 8; ++db) s += row[db];
    return s;
}

__global__ void __launch_bounds__(32)
stageB_boxfilter_wmma(const float* __restrict__ ws, float* __restrict__ T) {
    const int lane = threadIdx.x;           // 0..31, one wave
    const int m0   = blockIdx.x * 16;       // output row tile (i)
    const int n0   = blockIdx.y * 16;       // output col tile (j)
    const int p    = blockIdx.z;            // 0 = sum1 plane, 1 = sum2 plane
    const float* Cp = ws + p * (NQ * NB);

    const int half = lane >> 4;             // K-pair select (ISA 16x4 layout)
    const int ml   = lane & 15;             // M index (A) / N index (B,C,D)

    v8f acc = {0.f, 0.f, 0.f, 0.f, 0.f, 0.f, 0.f, 0.f};

    for (int k0 = 0; k0 < NQ; k0 += 4) {    // K loop over 62 chunks
        const int ka   = k0 + half * 2;     // K of .x ; K of .y is ka+1
        const int irow = m0 + ml;

        v2f a, b;
        // A = banded ones: W[irow, k] = (irow <= k <= irow+7)
        a.x = (ka     >= irow && ka     <= irow + 7) ? 1.0f : 0.0f;
        a.y = (ka + 1 >= irow && ka + 1 <= irow + 7) ? 1.0f : 0.0f;
        // B = horizontal 8-sum of C, guarded to 0 out of range
        b.x = hval(Cp, ka,     n0 + ml);
        b.y = hval(Cp, ka + 1, n0 + ml);

        acc = __builtin_amdgcn_wmma_f32_16x16x4_f32(
            false, a, false, b, (short)0, acc, false, false);
    }

    // C/D layout: VGPR v -> M = m0 + v + 8*half, N = n0 + (lane & 15)
#pragma unroll
    for (int v = 0; v < 8; ++v) {
        const int m = m0 + v + half * 8;
        const int n = n0 + ml;
        if (m < NH && n < NW) T[p * (NH * NW) + m * NW + n] = acc[v];
    }
}

// ---------------------------------------------------------------------------
// Stage C: argmin of (sum2 - sum1^2/n)/n over 4730 tiles (sqrt is monotone).
// First-index tie-break to match jnp.argmin.
// ---------------------------------------------------------------------------
__global__ void __launch_bounds__(256)
stageC_argmin(const float* __restrict__ T, int* __restrict__ loc) {
    const int tid = threadIdx.x;
    const float inv_n = 1.0f / (float)(PATCH * PATCH);

    float best = 3.402823466e38f;
    int   bidx = 0x7fffffff;
    for (int idx = tid; idx < NH * NW; idx += 256) {
        const float s1 = T[idx];
        const float s2 = T[NH * NW + idx];
        const float sc = (s2 - s1 * s1 * inv_n) * inv_n;
        if (sc < best || (sc == best && idx < bidx)) { best = sc; bidx = idx; }
    }

    __shared__ float bs[256];
    __shared__ int   bi[256];
    bs[tid] = best;
    bi[tid] = bidx;
    __syncthreads();
#pragma unroll
    for (int s = 128; s > 0; s >>= 1) {
        if (tid < s) {
            const float ob = bs[tid + s];
            const int   oi = bi[tid + s];
            if (ob < bs[tid] || (ob == bs[tid] && oi < bi[tid])) {
                bs[tid] = ob;
                bi[tid] = oi;
            }
        }
        __syncthreads();
    }
    if (tid == 0) {
        const int i = bi[0] / NW;
        const int j = bi[0] % NW;
        loc[0] = i * STEP;   // h0
        loc[1] = j * STEP;   // w0
    }
}

// ---------------------------------------------------------------------------
// Stage D: copy the 3x512x512 crop. w0 is 64-aligned -> float4-aligned.
// ---------------------------------------------------------------------------
__global__ void __launch_bounds__(128)
stageD_crop(const float* __restrict__ t, const int* __restrict__ loc,
            float* __restrict__ out) {
    const int row = blockIdx.x;    // 0..511
    const int ch  = blockIdx.y;    // 0..2
    const int tid = threadIdx.x;   // 0..127
    const int h0 = loc[0];
    const int w0 = loc[1];
    const float4* src = reinterpret_cast<const float4*>(
        t + (size_t)ch * PLANE + (size_t)(h0 + row) * IMG_W + w0);
    float4* dst = reinterpret_cast<float4*>(
        out + ((size_t)ch * PATCH + (size_t)row) * PATCH);
    dst[tid] = src[tid];
}

// ---------------------------------------------------------------------------
// Workspace layout (floats):
//   [0, 5766)          C1 (sum)      62*93
//   [5766, 11532)      C2 (sum^2)    62*93
//   [11532, 20992)     T  (2 planes) 2*55*86
//   [20992, 20994)     loc (2 ints)
// Total ~84 KB.
// ---------------------------------------------------------------------------
extern "C" void kernel_launch(void* const* d_in, const int* in_sizes, int n_in,
                              void* d_out, int out_size, void* d_ws, size_t ws_size,
                              hipStream_t stream) {
    (void)in_sizes; (void)n_in; (void)out_size; (void)ws_size;
    const float* t  = (const float*)d_in[0];
    float*       ws = (float*)d_ws;

    float* c1  = ws;
    float* c2  = ws + NQ * NB;
    float* T   = ws + 2 * NQ * NB;
    int*   loc = (int*)(ws + 2 * NQ * NB + 2 * NH * NW);

    stageA_blocksums<<<dim3(NB, NQ), 256, 0, stream>>>(t, c1, c2);
    stageB_boxfilter_wmma<<<dim3(4, 6, 2), 32, 0, stream>>>(ws, T);
    stageC_argmin<<<1, 256, 0, stream>>>(T, loc);
    stageD_crop<<<dim3(PATCH, IMG_C), 128, 0, stream>>>(t, loc, (float*)d_out);
}